// MultiScaleAttention_39359080300979
// MI455X (gfx1250) — compile-verified
//
#include <hip/hip_runtime.h>
#include <math.h>

typedef __attribute__((ext_vector_type(2))) float v2f;
typedef __attribute__((ext_vector_type(8))) float v8f;
typedef __attribute__((ext_vector_type(4))) unsigned int v4u;
typedef __attribute__((ext_vector_type(4))) int v4i;
typedef __attribute__((ext_vector_type(8))) int v8i;

#define BB 4
#define CC 512
#define LL 2048
#define OC 1536
#define HH 8
#define DD 64
#define GROUPS 32

// LDS row strides chosen so 2*stride == 16 (mod 64): the two 16-lane halves of
// a wave read disjoint bank halves (rows differ by 2 in the WMMA A/B layouts).
#define QS_STRIDE 136
#define KV_STRIDE 40   // 32 data dwords + 8 pad dwords (TDM pads in-flight)

#if defined(__AMDGCN__) && __has_builtin(__builtin_amdgcn_tensor_load_to_lds)
#define USE_TDM 1
#else
#define USE_TDM 0
#endif

__device__ __forceinline__ v8f wmma_f32(v2f a, v2f b, v8f c) {
    // V_WMMA_F32_16X16X4_F32 : D = A(16x4) * B(4x16) + C(16x16)
    return __builtin_amdgcn_wmma_f32_16x16x4_f32(
        false, a, false, b, (short)0, c, false, false);
}

#if USE_TDM
// TDM: load a 64(row) x 32(col) fp32 tile from a row-major [64][LL] tensor
// slice into LDS laid out [64][KV_STRIDE] (8 pad dwords appended per row).
__device__ __forceinline__ void tdm_load_64x32(float* lds_dst, const float* gsrc)
{
    unsigned long long ga = (unsigned long long)gsrc;
    unsigned int lds_lo = (unsigned int)(unsigned long long)lds_dst; // LDS offset
    v4u g0;
    g0.x = 1u;                                           // count=1, user mode
    g0.y = lds_lo;                                       // lds_addr
    g0.z = (unsigned int)(ga & 0xffffffffull);           // global_addr[31:0]
    g0.w = (unsigned int)((ga >> 32) & 0x1ffffffull)     // global_addr[56:32]
         | (2u << 30);                                   // type = 2 (image)
    v8i g1;
    g1[0] = (int)((2u << 16)      // data_size = 4B
                | (1u << 20)      // pad_enable
                | (4u << 22)      // pad_interval: every 32 dwords
                | (7u << 25));    // pad_amount: 8 dwords
    g1[1] = (int)((unsigned)LL << 16);   // tensor_dim0 = L (cols)
    g1[2] = (int)(64u << 16);            // tensor_dim1 = 64 (rows)
    g1[3] = (int)(32u << 16);            // tile_dim0 = 32
    g1[4] = 64;                          // tile_dim1 = 64 (tile_dim2 = 0)
    g1[5] = LL;                          // tensor_dim0_stride = L
    g1[6] = 0;
    g1[7] = 0;
    v4i z4 = {0, 0, 0, 0};
#if __clang_major__ >= 23
    v8i z8 = {0, 0, 0, 0, 0, 0, 0, 0};
    __builtin_amdgcn_tensor_load_to_lds(g0, g1, z4, z4, z8, 0);
#else
    __builtin_amdgcn_tensor_load_to_lds(g0, g1, z4, z4, 0);
#endif
}
#endif

// ---------------------------------------------------------------------------
// Kernel 1: qkv = clip(Wqkv @ x + bqkv).  Each wave -> 16(o) x 64(l) tile.
// ---------------------------------------------------------------------------
__global__ void __launch_bounds__(256) qkv_kernel(
    const float* __restrict__ x, const float* __restrict__ W,
    const float* __restrict__ bias,
    float* __restrict__ qb, float* __restrict__ kb, float* __restrict__ vb)
{
    const int wave = threadIdx.x >> 5;
    const int lane = threadIdx.x & 31;
    const int idx  = lane & 15;
    const int half = lane >> 4;

    const int ltiles = LL / 64;            // 32
    const int otiles = OC / 16;            // 96
    int gt = blockIdx.x * 8 + wave;
    int b  = gt / (otiles * ltiles);
    int r  = gt % (otiles * ltiles);
    int o0 = (r / ltiles) * 16;
    int l0 = (r % ltiles) * 64;

    const float* xb = x + (size_t)b * CC * LL;

    v8f acc[4] = {v8f{}, v8f{}, v8f{}, v8f{}};
    for (int c0 = 0; c0 < CC; c0 += 4) {
        v2f a;
        a.x = W[(size_t)(o0 + idx) * CC + c0 + 2 * half];
        a.y = W[(size_t)(o0 + idx) * CC + c0 + 2 * half + 1];
#pragma unroll
        for (int n = 0; n < 4; ++n) {
            v2f bm;
            bm.x = xb[(size_t)(c0 + 2 * half) * LL + l0 + n * 16 + idx];
            bm.y = xb[(size_t)(c0 + 2 * half + 1) * LL + l0 + n * 16 + idx];
            acc[n] = wmma_f32(a, bm, acc[n]);
        }
    }

    const int part = o0 / CC;                // 0=q 1=k 2=v (tile never crosses)
    float* dst = (part == 0) ? qb : (part == 1) ? kb : vb;
#pragma unroll
    for (int n = 0; n < 4; ++n) {
#pragma unroll
        for (int vv = 0; vv < 8; ++vv) {
            int o = o0 + vv + 8 * half;
            float val = acc[n][vv] + bias[o];
            val = fminf(fmaxf(val, -10.0f), 10.0f);
            dst[((size_t)b * CC + (o % CC)) * LL + l0 + n * 16 + idx] = val;
        }
    }
}

// ---------------------------------------------------------------------------
// Kernel 2: fused attention per (b,h).  Block = 256 thr (8 waves) handles 128
// query rows; K/V streamed through LDS in 64x32 tiles by the TDM.  Scores are
// clipped to [-10,10] before softmax => fixed max of 10, so exp(s-10) needs
// no flash-attention running-max rescale.
// ---------------------------------------------------------------------------
__global__ void __launch_bounds__(256) attn_kernel(
    const float* __restrict__ q, const float* __restrict__ k,
    const float* __restrict__ v, float* __restrict__ out)
{
    __shared__ float qs[64][QS_STRIDE];   // [d][i-local]
    __shared__ float ks[64][KV_STRIDE];   // [d][j-local]
    __shared__ float vs[64][KV_STRIDE];   // [d][j-local]
    __shared__ float ps[8][16][17];       // per-wave P staging (C->A layout)

    const int tid  = threadIdx.x;
    const int wave = tid >> 5;
    const int lane = tid & 31;
    const int idx  = lane & 15;
    const int half = lane >> 4;

    const int itiles = LL / 128;             // 16
    const int bh = blockIdx.x / itiles;      // b*H + h
    const int i0 = (blockIdx.x % itiles) * 128;
    const int iw = wave * 16;

    const float* qp = q + (size_t)bh * DD * LL;
    const float* kp = k + (size_t)bh * DD * LL;
    const float* vp = v + (size_t)bh * DD * LL;
    float* op = out + (size_t)bh * DD * LL;

    for (int e = tid; e < 64 * 128; e += 256) {
        int d = e >> 7, i = e & 127;
        qs[d][i] = qp[(size_t)d * LL + i0 + i];
    }

    const float scale = 0.125f;              // D^-0.5 = 1/8
    v8f oacc[4] = {v8f{}, v8f{}, v8f{}, v8f{}};
    float rs[8] = {0.f, 0.f, 0.f, 0.f, 0.f, 0.f, 0.f, 0.f};

    for (int j0 = 0; j0 < LL; j0 += 32) {
        __syncthreads();                     // prior reads of ks/vs done
#if USE_TDM
        if (wave == 0) {
            tdm_load_64x32(&ks[0][0], kp + j0);
            tdm_load_64x32(&vs[0][0], vp + j0);
            __builtin_amdgcn_s_wait_tensorcnt(0);
        }
#else
        for (int e = tid; e < 64 * 32; e += 256) {
            int d = e >> 5, j = e & 31;
            ks[d][j] = kp[(size_t)d * LL + j0 + j];
            vs[d][j] = vp[(size_t)d * LL + j0 + j];
        }
#endif
        __syncthreads();                     // tiles visible to all waves

#pragma unroll
        for (int jt = 0; jt < 32; jt += 16) {
            // S tile: 16(i) x 16(j), K-dim = d (64) in steps of 4
            v8f s = {};
#pragma unroll
            for (int dd = 0; dd < DD; dd += 4) {
                v2f a, bm;
                a.x  = qs[dd + 2 * half][iw + idx];
                a.y  = qs[dd + 2 * half + 1][iw + idx];
                bm.x = ks[dd + 2 * half][jt + idx];
                bm.y = ks[dd + 2 * half + 1][jt + idx];
                s = wmma_f32(a, bm, s);
            }

            // p = exp(clip(s*scale) - 10); stage P for A-layout re-read
#pragma unroll
            for (int vv = 0; vv < 8; ++vv) {
                float t = fminf(fmaxf(s[vv] * scale, -10.0f), 10.0f);
                float p = __expf(t - 10.0f);
                rs[vv] += p;
                ps[wave][vv + 8 * half][idx] = p;
            }

            // O += P * V^T : M=i, N=d, K=j(16)
#pragma unroll
            for (int jj = 0; jj < 16; jj += 4) {
                v2f a;
                a.x = ps[wave][idx][jj + 2 * half];
                a.y = ps[wave][idx][jj + 2 * half + 1];
#pragma unroll
                for (int n = 0; n < 4; ++n) {
                    v2f bm;
                    bm.x = vs[n * 16 + idx][jt + jj + 2 * half];
                    bm.y = vs[n * 16 + idx][jt + jj + 2 * half + 1];
                    oacc[n] = wmma_f32(a, bm, oacc[n]);
                }
            }
        }
    }

    // Z[i]: reduce partial row sums across the 16 lanes of each half
#pragma unroll
    for (int m = 1; m < 16; m <<= 1) {
#pragma unroll
        for (int vv = 0; vv < 8; ++vv)
            rs[vv] += __shfl_xor(rs[vv], m, 32);
    }

#pragma unroll
    for (int n = 0; n < 4; ++n) {
#pragma unroll
        for (int vv = 0; vv < 8; ++vv) {
            float val = oacc[n][vv] / rs[vv];
            op[(size_t)(n * 16 + idx) * LL + i0 + iw + vv + 8 * half] = val;
        }
    }
}

// ---------------------------------------------------------------------------
// Kernel 3: z = Wproj @ attnout + bproj + x, plus per-(b,group) sum/sumsq.
// Each 16-row tile lies entirely inside one GroupNorm group (16 channels).
// ---------------------------------------------------------------------------
__global__ void __launch_bounds__(256) proj_kernel(
    const float* __restrict__ ain, const float* __restrict__ W,
    const float* __restrict__ bias, const float* __restrict__ x,
    float* __restrict__ z, float* __restrict__ gstats)
{
    const int wave = threadIdx.x >> 5;
    const int lane = threadIdx.x & 31;
    const int idx  = lane & 15;
    const int half = lane >> 4;

    const int ltiles = LL / 64;            // 32
    const int otiles = CC / 16;            // 32
    int gt = blockIdx.x * 8 + wave;
    int b  = gt / (otiles * ltiles);
    int r  = gt % (otiles * ltiles);
    int o0 = (r / ltiles) * 16;
    int l0 = (r % ltiles) * 64;

    const float* ab = ain + (size_t)b * CC * LL;
    const float* xb = x   + (size_t)b * CC * LL;

    v8f acc[4] = {v8f{}, v8f{}, v8f{}, v8f{}};
    for (int c0 = 0; c0 < CC; c0 += 4) {
        v2f a;
        a.x = W[(size_t)(o0 + idx) * CC + c0 + 2 * half];
        a.y = W[(size_t)(o0 + idx) * CC + c0 + 2 * half + 1];
#pragma unroll
        for (int n = 0; n < 4; ++n) {
            v2f bm;
            bm.x = ab[(size_t)(c0 + 2 * half) * LL + l0 + n * 16 + idx];
            bm.y = ab[(size_t)(c0 + 2 * half + 1) * LL + l0 + n * 16 + idx];
            acc[n] = wmma_f32(a, bm, acc[n]);
        }
    }

    float s1 = 0.f, s2 = 0.f;
#pragma unroll
    for (int n = 0; n < 4; ++n) {
#pragma unroll
        for (int vv = 0; vv < 8; ++vv) {
            int o = o0 + vv + 8 * half;
            size_t off = (size_t)o * LL + l0 + n * 16 + idx;
            float val = acc[n][vv] + bias[o] + xb[off];
            z[(size_t)b * CC * LL + off] = val;
            s1 += val;
            s2 += val * val;
        }
    }
#pragma unroll
    for (int m = 1; m < 32; m <<= 1) {
        s1 += __shfl_xor(s1, m, 32);
        s2 += __shfl_xor(s2, m, 32);
    }
    if (lane == 0) {
        int g = o0 >> 4;                 // group = channel/16
        atomicAdd(&gstats[(b * GROUPS + g) * 2 + 0], s1);
        atomicAdd(&gstats[(b * GROUPS + g) * 2 + 1], s2);
    }
}

// ---------------------------------------------------------------------------
// Kernel 4: GroupNorm apply -> d_out
// ---------------------------------------------------------------------------
__global__ void __launch_bounds__(256) gn_kernel(
    const float* __restrict__ z, const float* __restrict__ gstats,
    const float* __restrict__ gamma, const float* __restrict__ beta,
    float* __restrict__ out)
{
    int t = blockIdx.x * blockDim.x + threadIdx.x;     // one float4 each
    int e = t * 4;
    int b = e / (CC * LL);
    int rem = e % (CC * LL);
    int c = rem / LL;
    int g = c >> 4;

    float s1 = gstats[(b * GROUPS + g) * 2 + 0];
    float s2 = gstats[(b * GROUPS + g) * 2 + 1];
    const float invN = 1.0f / (float)((CC / GROUPS) * LL);   // 1/32768
    float mu   = s1 * invN;
    float var  = s2 * invN - mu * mu;
    float rstd = rsqrtf(var + 1e-5f);
    float ga = gamma[c], be = beta[c];

    const float4 zv = *(const float4*)(z + e);
    float4 ov;
    ov.x = (zv.x - mu) * rstd * ga + be;
    ov.y = (zv.y - mu) * rstd * ga + be;
    ov.z = (zv.z - mu) * rstd * ga + be;
    ov.w = (zv.w - mu) * rstd * ga + be;
    *(float4*)(out + e) = ov;
}

// ---------------------------------------------------------------------------
extern "C" void kernel_launch(void* const* d_in, const int* in_sizes, int n_in,
                              void* d_out, int out_size, void* d_ws, size_t ws_size,
                              hipStream_t stream)
{
    const float* x     = (const float*)d_in[0];
    const float* Wqkv  = (const float*)d_in[1];
    const float* bqkv  = (const float*)d_in[2];
    const float* Wproj = (const float*)d_in[3];
    const float* bproj = (const float*)d_in[4];
    const float* gamma = (const float*)d_in[5];
    const float* beta  = (const float*)d_in[6];
    float* out = (float*)d_out;

    const size_t n = (size_t)BB * CC * LL;     // 4,194,304 floats
    float* ws = (float*)d_ws;
    float* qb = ws;
    float* kb = ws + n;
    float* vb = ws + 2 * n;
    float* ao = ws + 3 * n;
    float* zb = ws + 4 * n;
    float* gstats = ws + 5 * n;

    hipMemsetAsync(gstats, 0, BB * GROUPS * 2 * sizeof(float), stream);

    qkv_kernel <<<(BB * (OC/16) * (LL/64)) / 8, 256, 0, stream>>>(x, Wqkv, bqkv, qb, kb, vb);
    attn_kernel<<<BB * HH * (LL/128),          256, 0, stream>>>(qb, kb, vb, ao);
    proj_kernel<<<(BB * (CC/16) * (LL/64)) / 8, 256, 0, stream>>>(ao, Wproj, bproj, x, zb, gstats);
    gn_kernel  <<<(int)(n / 4 / 256),           256, 0, stream>>>(zb, gstats, gamma, beta, out);
}